// GraphDecoderEVRP_57011395887687
// MI455X (gfx1250) — compile-verified
//
#include <hip/hip_runtime.h>
#include <math.h>

#define B_  256
#define N_  2048
#define D_  128
#define H_  8
#define HD_ 16

typedef __attribute__((ext_vector_type(2))) float v2f;
typedef __attribute__((ext_vector_type(4))) float v4f;
typedef __attribute__((ext_vector_type(8))) float v8f;

static __device__ __forceinline__ v4f nt_load4(const float* p) {
    return __builtin_nontemporal_load((const v4f*)p);
}

// ---------------------------------------------------------------------------
// Kernel 1: query[b,:] = ctx_proj[b,:] + ctx[b,:] @ W_ctx   (16 batches/wave)
// ctx[b, 0:128]   = node_emb[b, first_a[b], :]
// ctx[b, 128:256] = node_emb[b, current_node[b], :]
// Uses V_WMMA_F32_16X16X4_F32: D(16x16) += A(16x4) * B(4x16)
// ---------------------------------------------------------------------------
__global__ __launch_bounds__(32) void k_query(
    const float* __restrict__ ctx_proj,   // [B,128]
    const float* __restrict__ node_emb,   // [B,N,128]
    const float* __restrict__ W_ctx,      // [256,128]
    const int*   __restrict__ first_a,    // [B]
    const int*   __restrict__ cur_node,   // [B]
    float*       __restrict__ q_ws)       // [B,128]
{
    const int lane = threadIdx.x;
    const int m    = lane & 15;     // A-matrix row / D column
    const int half = lane >> 4;     // selects K pair (A) / row offset (D)
    const int b0   = blockIdx.x * 16;
    const int bidx = b0 + m;

    const int fa = first_a[bidx];
    const int cn = cur_node[bidx];
    const float* rowA = node_emb + ((size_t)bidx * N_ + fa) * D_;  // ctx cols 0..127
    const float* rowB = node_emb + ((size_t)bidx * N_ + cn) * D_;  // ctx cols 128..255

    for (int jt = 0; jt < 8; ++jt) {
        const int ncol = jt * 16 + m;
        v8f c = {};
        for (int kk = 0; kk < 64; ++kk) {
            const int kb = kk * 4 + half * 2;   // this lane's K pair
            const float* ap = (kb < 128) ? (rowA + kb) : (rowB + (kb - 128));
            v2f a = *(const v2f*)ap;
            v2f bf;
            bf.x = W_ctx[(size_t)kb * D_ + ncol];
            bf.y = W_ctx[(size_t)(kb + 1) * D_ + ncol];
            c = __builtin_amdgcn_wmma_f32_16x16x4_f32(
                    false, a, false, bf, (short)0, c, false, false);
        }
        #pragma unroll
        for (int r = 0; r < 8; ++r) {
            const int ob = b0 + r + half * 8;   // D row = batch within group
            q_ws[(size_t)ob * D_ + ncol] = c[r] + ctx_proj[(size_t)ob * D_ + ncol];
        }
    }
}

// ---------------------------------------------------------------------------
// Kernel 2: per (b,h) flash-style masked attention over N keys of dim 16.
// Streams glimpse_K / glimpse_V once with NT b128 loads (bandwidth bound).
// ---------------------------------------------------------------------------
__global__ __launch_bounds__(256) void k_attn(
    const float* __restrict__ q_ws,   // [B,128]
    const float* __restrict__ gK,     // [H,B,N,16]
    const float* __restrict__ gV,     // [H,B,N,16]
    const unsigned char* __restrict__ mask,  // [B,N]
    float*       __restrict__ heads_ws)      // [B,128]  (h-major within row)
{
    const int b = blockIdx.x, h = blockIdx.y, t = threadIdx.x;

    __shared__ float q_s[HD_];
    __shared__ float m_s[256], l_s[256];
    __shared__ float acc_s[256][HD_];

    if (t < HD_) q_s[t] = q_ws[(size_t)b * D_ + h * HD_ + t];
    __syncthreads();

    float q[HD_];
    #pragma unroll
    for (int k = 0; k < HD_; ++k) q[k] = q_s[k];

    const float* Kb = gK + ((size_t)(h * B_ + b) * N_) * HD_;
    const float* Vb = gV + ((size_t)(h * B_ + b) * N_) * HD_;
    const unsigned char* mrow = mask + (size_t)b * N_;

    float mv = -INFINITY, l = 0.f;
    float acc[HD_];
    #pragma unroll
    for (int k = 0; k < HD_; ++k) acc[k] = 0.f;

    for (int n = t; n < N_; n += 256) {
        if (mrow[n]) continue;
        const float* Kr = Kb + (size_t)n * HD_;
        v4f k0 = nt_load4(Kr), k1 = nt_load4(Kr + 4),
            k2 = nt_load4(Kr + 8), k3 = nt_load4(Kr + 12);
        float s = q[0]*k0.x + q[1]*k0.y + q[2]*k0.z + q[3]*k0.w
                + q[4]*k1.x + q[5]*k1.y + q[6]*k1.z + q[7]*k1.w
                + q[8]*k2.x + q[9]*k2.y + q[10]*k2.z + q[11]*k2.w
                + q[12]*k3.x + q[13]*k3.y + q[14]*k3.z + q[15]*k3.w;
        s *= 0.25f;                                   // 1/sqrt(16)
        if (s > mv) {
            float sc = __expf(mv - s);
            l *= sc;
            #pragma unroll
            for (int k = 0; k < HD_; ++k) acc[k] *= sc;
            mv = s;
        }
        float w = __expf(s - mv);
        l += w;
        const float* Vr = Vb + (size_t)n * HD_;
        v4f v0 = nt_load4(Vr), v1 = nt_load4(Vr + 4),
            v2 = nt_load4(Vr + 8), v3 = nt_load4(Vr + 12);
        acc[0] += w*v0.x;  acc[1] += w*v0.y;  acc[2] += w*v0.z;  acc[3] += w*v0.w;
        acc[4] += w*v1.x;  acc[5] += w*v1.y;  acc[6] += w*v1.z;  acc[7] += w*v1.w;
        acc[8] += w*v2.x;  acc[9] += w*v2.y;  acc[10]+= w*v2.z;  acc[11]+= w*v2.w;
        acc[12]+= w*v3.x;  acc[13]+= w*v3.y;  acc[14]+= w*v3.z;  acc[15]+= w*v3.w;
    }

    m_s[t] = mv; l_s[t] = l;
    #pragma unroll
    for (int k = 0; k < HD_; ++k) acc_s[t][k] = acc[k];
    __syncthreads();

    for (int str = 128; str > 0; str >>= 1) {
        if (t < str) {
            float l2 = l_s[t + str];
            if (l2 > 0.f) {
                float m2 = m_s[t + str];
                float m1 = m_s[t], l1 = l_s[t];
                if (l1 <= 0.f) {
                    m_s[t] = m2; l_s[t] = l2;
                    #pragma unroll
                    for (int k = 0; k < HD_; ++k) acc_s[t][k] = acc_s[t + str][k];
                } else {
                    float mn = fmaxf(m1, m2);
                    float a1 = __expf(m1 - mn), a2 = __expf(m2 - mn);
                    m_s[t] = mn;
                    l_s[t] = l1 * a1 + l2 * a2;
                    #pragma unroll
                    for (int k = 0; k < HD_; ++k)
                        acc_s[t][k] = acc_s[t][k] * a1 + acc_s[t + str][k] * a2;
                }
            }
        }
        __syncthreads();
    }

    if (t < HD_)
        heads_ws[(size_t)b * D_ + h * HD_ + t] = acc_s[0][t] / l_s[0];
}

// ---------------------------------------------------------------------------
// Kernel 3: glimpse = heads(16x128) @ W_out(128x128) via WMMA f32 16x16x4.
// ---------------------------------------------------------------------------
__global__ __launch_bounds__(32) void k_glimpse(
    const float* __restrict__ heads_ws,  // [B,128]
    const float* __restrict__ W_out,     // [128,128]
    float*       __restrict__ g_ws)      // [B,128]
{
    const int lane = threadIdx.x;
    const int m    = lane & 15;
    const int half = lane >> 4;
    const int b0   = blockIdx.x * 16;
    const float* rowA = heads_ws + (size_t)(b0 + m) * D_;

    for (int jt = 0; jt < 8; ++jt) {
        const int ncol = jt * 16 + m;
        v8f c = {};
        for (int kk = 0; kk < 32; ++kk) {
            const int kb = kk * 4 + half * 2;
            v2f a = *(const v2f*)(rowA + kb);
            v2f bf;
            bf.x = W_out[(size_t)kb * D_ + ncol];
            bf.y = W_out[(size_t)(kb + 1) * D_ + ncol];
            c = __builtin_amdgcn_wmma_f32_16x16x4_f32(
                    false, a, false, bf, (short)0, c, false, false);
        }
        #pragma unroll
        for (int r = 0; r < 8; ++r) {
            const int ob = b0 + r + half * 8;
            g_ws[(size_t)ob * D_ + ncol] = c[r];
        }
    }
}

// ---------------------------------------------------------------------------
// Kernel 4: per b: logits = tanh(glimpse . logit_K / sqrt(128)) * 10,
// mask -> -inf, then log_softmax over N. Streams logit_K once (NT).
// ---------------------------------------------------------------------------
__global__ __launch_bounds__(256) void k_logits(
    const float* __restrict__ g_ws,      // [B,128]
    const float* __restrict__ logit_K,   // [B,N,128]
    const unsigned char* __restrict__ mask,  // [B,N]
    float*       __restrict__ out)       // [B,N]
{
    const int b = blockIdx.x, t = threadIdx.x;
    __shared__ float gl[D_];
    __shared__ float lg[N_];
    __shared__ float red[256];

    if (t < D_) gl[t] = g_ws[(size_t)b * D_ + t];
    __syncthreads();

    const float* Kb = logit_K + (size_t)b * N_ * D_;
    const unsigned char* mrow = mask + (size_t)b * N_;
    const float inv_sqrt_d = 0.08838834764831845f;   // 1/sqrt(128)

    for (int n = t; n < N_; n += 256) {
        float s;
        if (mrow[n]) {
            s = -INFINITY;
        } else {
            const float* r = Kb + (size_t)n * D_;
            float acc = 0.f;
            #pragma unroll 4
            for (int k = 0; k < D_; k += 4) {
                v4f v = nt_load4(r + k);
                acc += gl[k]*v.x + gl[k+1]*v.y + gl[k+2]*v.z + gl[k+3]*v.w;
            }
            s = tanhf(acc * inv_sqrt_d) * 10.0f;
        }
        lg[n] = s;
    }
    __syncthreads();

    // block max
    float mx = -INFINITY;
    for (int n = t; n < N_; n += 256) mx = fmaxf(mx, lg[n]);
    red[t] = mx;
    __syncthreads();
    for (int str = 128; str > 0; str >>= 1) {
        if (t < str) red[t] = fmaxf(red[t], red[t + str]);
        __syncthreads();
    }
    mx = red[0];
    __syncthreads();

    // block sum of exp
    float sm = 0.f;
    for (int n = t; n < N_; n += 256) sm += __expf(lg[n] - mx);
    red[t] = sm;
    __syncthreads();
    for (int str = 128; str > 0; str >>= 1) {
        if (t < str) red[t] += red[t + str];
        __syncthreads();
    }
    const float logZ = mx + __logf(red[0]);

    for (int n = t; n < N_; n += 256)
        out[(size_t)b * N_ + n] = lg[n] - logZ;   // masked: -inf - logZ = -inf
}

// ---------------------------------------------------------------------------
extern "C" void kernel_launch(void* const* d_in, const int* in_sizes, int n_in,
                              void* d_out, int out_size, void* d_ws, size_t ws_size,
                              hipStream_t stream) {
    const float* ctx_proj = (const float*)d_in[0];   // [B,1,128]
    const float* node_emb = (const float*)d_in[1];   // [B,N,128]
    const float* gK       = (const float*)d_in[2];   // [H,B,1,N,16]
    const float* gV       = (const float*)d_in[3];   // [H,B,1,N,16]
    const float* logit_K  = (const float*)d_in[4];   // [B,1,N,128]
    const float* W_ctx    = (const float*)d_in[5];   // [256,128]
    const float* W_out    = (const float*)d_in[6];   // [128,128]
    const int*   first_a  = (const int*)d_in[7];     // [B,1]
    const int*   cur_node = (const int*)d_in[8];     // [B,1]
    const unsigned char* mask = (const unsigned char*)d_in[9]; // [B,1,N] bool
    float* out = (float*)d_out;                      // [B,1,N]

    float* q_ws     = (float*)d_ws;                  // [B,128]
    float* heads_ws = q_ws + B_ * D_;                // [B,128]
    float* g_ws     = heads_ws + B_ * D_;            // [B,128]

    k_query  <<<dim3(B_ / 16), dim3(32),  0, stream>>>(ctx_proj, node_emb, W_ctx,
                                                       first_a, cur_node, q_ws);
    k_attn   <<<dim3(B_, H_),  dim3(256), 0, stream>>>(q_ws, gK, gV, mask, heads_ws);
    k_glimpse<<<dim3(B_ / 16), dim3(32),  0, stream>>>(heads_ws, W_out, g_ws);
    k_logits <<<dim3(B_),      dim3(256), 0, stream>>>(g_ws, logit_K, mask, out);
}